// GroupedPNMLP_6588479832305
// MI455X (gfx1250) — compile-verified
//
#include <hip/hip_runtime.h>

typedef float v2f __attribute__((ext_vector_type(2)));
typedef float v8f __attribute__((ext_vector_type(8)));

#define N_SAMPLES 131072
#define N_NODES   24
#define ROW_F     (N_NODES * 32)          // floats per sample row of h
#define SAMPLES_PER_WAVE  64
#define SAMPLES_PER_BLOCK 512

// pos_of_node[n]: flat (g*4+k) slot in GROUPING whose node id == n
__constant__ int c_pos[24] = {0,4,8,1,5,9,2,6,10,3,7,11,12,13,14,15,16,20,17,21,18,22,19,23};
// GROUPING rows (members of each group), for the validity mask
__constant__ int c_gmem[6][4] = {{0,3,6,9},{1,4,7,10},{2,5,8,11},
                                 {12,13,14,15},{16,18,20,22},{17,19,21,23}};

__global__ __launch_bounds__(256)
void grouped_pn_mlp_kernel(const float* __restrict__ h,
                           const int*   __restrict__ valid,
                           const float* __restrict__ W1,
                           const float* __restrict__ b1,
                           const float* __restrict__ W2,
                           const float* __restrict__ b2,
                           float* __restrict__ out)
{
    const int lane = threadIdx.x & 31;
    const int wave = threadIdx.x >> 5;
    const int node = blockIdx.y;

    const int p   = c_pos[node];       // weight slot
    const int g   = p >> 2;            // group id
    const int lo  = lane & 15;
    const int hiK = (lane >= 16) ? 2 : 0;   // A/B fragment K split per ISA layout

    const float* W1p = W1 + (size_t)p * 1024;   // 32x32, row = in_ch, col = hid
    const float* b1p = b1 + p * 32;
    const float* W2p = W2 + p * 32;
    const float  b2v = b2[p];

    // ---- Layer-1 weight B-fragments: B is K(32) x N(32), two 16-wide N tiles.
    // Per 16x16x4 WMMA step s: VGPR0 holds K = 4s + {0 | 2}, VGPR1 holds K = 4s + {1 | 3}.
    v2f wf[8][2];
#pragma unroll
    for (int s = 0; s < 8; ++s) {
#pragma unroll
        for (int t = 0; t < 2; ++t) {
            const int n = t * 16 + lo;
            wf[s][t].x = W1p[(4 * s + 0 + hiK) * 32 + n];
            wf[s][t].y = W1p[(4 * s + 1 + hiK) * 32 + n];
        }
    }

    const float b1a = b1p[lo];          // bias for N-tile 0 (this lane's column)
    const float b1b = b1p[16 + lo];     // bias for N-tile 1
    const float w2a = W2p[lo];          // layer-2 weight for this lane's column
    const float w2b = W2p[16 + lo];

    const int tile0 = blockIdx.x * SAMPLES_PER_BLOCK + wave * SAMPLES_PER_WAVE;

#pragma unroll
    for (int sub = 0; sub < 4; ++sub) {
        const int tb = tile0 + sub * 16;
        // A-fragment source: lane pair (l, l+16) covers 16 contiguous bytes of
        // one sample row; 8 K-steps consume the whole 128B row.
        const float* rowp = h + (size_t)(tb + lo) * ROW_F + node * 32 + hiK;

        if (sub < 3)
            __builtin_prefetch(rowp + (size_t)16 * ROW_F, 0, 1); // next sub-tile

        // C init = bias (broadcast along M; lane fixes N, regs span M)
        v8f acc0, acc1;
#pragma unroll
        for (int r = 0; r < 8; ++r) { acc0[r] = b1a; acc1[r] = b1b; }

#pragma unroll
        for (int s = 0; s < 8; ++s) {
            v2f a = *(const v2f*)(rowp + 4 * s);     // global_load_b64, 8B aligned
            a.x = fmaxf(a.x, 0.0f);                  // relu(h) feeds layer 1
            a.y = fmaxf(a.y, 0.0f);
            acc0 = __builtin_amdgcn_wmma_f32_16x16x4_f32(
                false, a, false, wf[s][0], (short)0, acc0, false, false);
            acc1 = __builtin_amdgcn_wmma_f32_16x16x4_f32(
                false, a, false, wf[s][1], (short)0, acc1, false, false);
        }

        // ---- Layer 2: relu, scale each lane's column by W2[n], reduce over N.
        // D layout: reg r / lane l -> element [M = r + 8*(l>>4)][N = l%16 (+16 for tile1)]
        float srow[8];
#pragma unroll
        for (int r = 0; r < 8; ++r)
            srow[r] = fmaxf(acc0[r], 0.0f) * w2a + fmaxf(acc1[r], 0.0f) * w2b;

        // butterfly sum across the 16-lane half (masks < 16 stay inside a half)
#pragma unroll
        for (int off = 1; off < 16; off <<= 1) {
#pragma unroll
            for (int r = 0; r < 8; ++r)
                srow[r] += __shfl_xor(srow[r], off, 32);
        }

        // group-valid mask: lane `lo` owns sample tb+lo
        const int vb = (tb + lo) * N_NODES;
        const int vs = valid[vb + c_gmem[g][0]] + valid[vb + c_gmem[g][1]]
                     + valid[vb + c_gmem[g][2]] + valid[vb + c_gmem[g][3]];
        const float gv = (vs > 0) ? 1.0f : 0.0f;

        // 16 writer lanes: lanes 0..7 -> rows 0..7, lanes 16..23 -> rows 8..15
        const int row = (lane & 7) + ((lane >> 4) << 3);
        float v = srow[0];
#pragma unroll
        for (int r = 1; r < 8; ++r)
            v = ((lane & 7) == r) ? srow[r] : v;
        const float gvr = __shfl(gv, row, 32);

        if ((lane & 15) < 8)
            out[(size_t)(tb + row) * N_NODES + node] = (v + b2v) * gvr;
    }
}

extern "C" void kernel_launch(void* const* d_in, const int* in_sizes, int n_in,
                              void* d_out, int out_size, void* d_ws, size_t ws_size,
                              hipStream_t stream) {
    (void)in_sizes; (void)n_in; (void)out_size; (void)d_ws; (void)ws_size;
    const float* h     = (const float*)d_in[0];
    const int*   valid = (const int*)  d_in[1];
    const float* W1    = (const float*)d_in[2];
    const float* b1    = (const float*)d_in[3];
    const float* W2    = (const float*)d_in[4];
    const float* b2    = (const float*)d_in[5];
    float* out = (float*)d_out;

    dim3 grid(N_SAMPLES / SAMPLES_PER_BLOCK, N_NODES);
    dim3 block(256);
    grouped_pn_mlp_kernel<<<grid, block, 0, stream>>>(h, valid, W1, b1, W2, b2, out);
}